// EpropLSTM_29575144800638
// MI455X (gfx1250) — compile-verified
//
#include <hip/hip_runtime.h>
#include <math.h>

typedef __attribute__((ext_vector_type(2))) float v2f;
typedef __attribute__((ext_vector_type(4))) float f4;
typedef __attribute__((ext_vector_type(8))) float v8f;

#define BATCH 16
#define HID   1024
#define IN    512
#define NBH   (BATCH * HID)   // 16384

// d_out flat offsets (floats), in reference return order
#define OFF_H     0
#define OFF_C     16384
#define OFF_EWIX  32768
#define OFF_EWIH  8421376
#define OFF_EBI   25198592
#define OFF_EWFX  25214976
#define OFF_EWFH  33603584
#define OFF_EBF   50380800
#define OFF_EWCX  50397184
#define OFF_EWCH  58785792
#define OFF_EBC   75563008

// Workspace layout (floats): gates i,f,o,ch then coefs a_i,a_f,a_c (7 * 16384)
#define WS_GATES 0
#define WS_COEF  (4 * NBH)

// ---------------------------------------------------------------------------
// Kernel A: gate pre-activations via fp32 WMMA (exact), + bias + sigmoid/tanh.
// One wave (32 threads) per 16x16 (batch x H) tile per gate.
// A: 16x4 fp32 tile of x / h_last; B: 4x16 tile of W^T (i.e. rows of W).
// Lane mapping per ISA fp32 layouts: lanes 0-15 <-> K={0,1}, lanes 16-31 <-> K={2,3}.
// ---------------------------------------------------------------------------
__global__ __launch_bounds__(32) void gates_wmma(
    const float* __restrict__ x, const float* __restrict__ h_last,
    const float* __restrict__ w_ix, const float* __restrict__ w_ih, const float* __restrict__ b_i,
    const float* __restrict__ w_fx, const float* __restrict__ w_fh, const float* __restrict__ b_f,
    const float* __restrict__ w_ox, const float* __restrict__ w_oh, const float* __restrict__ b_o,
    const float* __restrict__ w_cx, const float* __restrict__ w_ch, const float* __restrict__ b_c,
    float* __restrict__ ws)
{
  const int g  = blockIdx.y;        // 0=i, 1=f, 2=o, 3=c_hat
  const int h0 = blockIdx.x * 16;   // H tile base

  const float* wx = w_ix; const float* wh = w_ih; const float* bias = b_i;
  if (g == 1) { wx = w_fx; wh = w_fh; bias = b_f; }
  else if (g == 2) { wx = w_ox; wh = w_oh; bias = b_o; }
  else if (g == 3) { wx = w_cx; wh = w_ch; bias = b_c; }

  const int lane = threadIdx.x;         // 0..31, EXEC all-ones (required for WMMA)
  const int mn   = lane & 15;           // M for A-side, N for B-side
  const int koff = (lane >> 4) * 2;     // K sub-offset per half-wave

  v8f acc = {};

  // x @ wx^T : A row = x[batch mn], B col = wx row (h0+mn)
  {
    const float* arow = x  + mn * IN;
    const float* brow = wx + (h0 + mn) * IN;
    for (int k = 0; k < IN; k += 4) {
      v2f a = *(const v2f*)(arow + k + koff);
      v2f b = *(const v2f*)(brow + k + koff);
      acc = __builtin_amdgcn_wmma_f32_16x16x4_f32(
          false, a, false, b, (short)0, acc, false, false);
    }
  }
  // h_last @ wh^T
  {
    const float* arow = h_last + mn * HID;
    const float* brow = wh + (h0 + mn) * HID;
    for (int k = 0; k < HID; k += 4) {
      v2f a = *(const v2f*)(arow + k + koff);
      v2f b = *(const v2f*)(brow + k + koff);
      acc = __builtin_amdgcn_wmma_f32_16x16x4_f32(
          false, a, false, b, (short)0, acc, false, false);
    }
  }

  // D layout: VGPR r -> M = r + 8*(lane>>4); N = lane&15
  const int   n  = h0 + mn;
  const float bv = bias[n];
  float* outg = ws + WS_GATES + g * NBH;
  const int mhi = (lane >> 4) * 8;
#pragma unroll
  for (int r = 0; r < 8; ++r) {
    const int m = r + mhi;
    float z = acc[r] + bv;
    float v = (g == 3) ? tanhf(z) : (1.0f / (1.0f + expf(-z)));
    outg[m * HID + n] = v;
  }
}

// ---------------------------------------------------------------------------
// Kernel B: combine gates -> h, c, bias traces, and broadcast coefficients.
// ---------------------------------------------------------------------------
__global__ __launch_bounds__(256) void combine(
    const float* __restrict__ c_last,
    const float* __restrict__ e_b_i, const float* __restrict__ e_b_f,
    const float* __restrict__ e_b_c,
    float* __restrict__ ws, float* __restrict__ out)
{
  const int t = blockIdx.x * blockDim.x + threadIdx.x;
  if (t >= NBH) return;

  const float i  = ws[WS_GATES + 0 * NBH + t];
  const float f  = ws[WS_GATES + 1 * NBH + t];
  const float o  = ws[WS_GATES + 2 * NBH + t];
  const float ch = ws[WS_GATES + 3 * NBH + t];
  const float cl = c_last[t];

  const float c = f * cl + i * ch;
  const float h = o * c;

  const float di  = i * (1.0f - i);
  const float df  = f * (1.0f - f);
  const float dch = 1.0f - ch * ch;

  const float a_i = di * ch;
  const float a_f = df * cl;
  const float a_c = dch * i;

  ws[WS_COEF + 0 * NBH + t] = a_i;
  ws[WS_COEF + 1 * NBH + t] = a_f;
  ws[WS_COEF + 2 * NBH + t] = a_c;

  out[OFF_H   + t] = h;
  out[OFF_C   + t] = c;
  out[OFF_EBI + t] = e_b_i[t] * f + a_i;   // di*c_hat == a_i
  out[OFF_EBF + t] = e_b_f[t] * f + a_f;   // df*c_last == a_f
  out[OFF_EBC + t] = e_b_c[t] * f + dch * i;
}

// ---------------------------------------------------------------------------
// Kernel C: stream x-side traces (K=512). One block (128 thr) per (b,h) row,
// grid.y selects {i,f,c}. float4 loads/stores; f and a are wave-uniform.
// ---------------------------------------------------------------------------
__global__ __launch_bounds__(128) void trace_x(
    const float* __restrict__ x,
    const float* __restrict__ e_ix, const float* __restrict__ e_fx,
    const float* __restrict__ e_cx,
    const float* __restrict__ ws, float* __restrict__ out)
{
  const int bh    = blockIdx.x;     // 0..16383
  const int which = blockIdx.y;     // 0..2
  const int b     = bh >> 10;       // H == 1024

  const float f = ws[WS_GATES + 1 * NBH + bh];
  const float a = ws[WS_COEF + which * NBH + bh];

  const float* e = e_ix; float* dst = out + OFF_EWIX;
  if (which == 1)      { e = e_fx; dst = out + OFF_EWFX; }
  else if (which == 2) { e = e_cx; dst = out + OFF_EWCX; }

  const int base = bh * IN + (threadIdx.x << 2);
  const f4 ev = *(const f4*)(e + base);
  const f4 xv = *(const f4*)(x + b * IN + (threadIdx.x << 2));
  f4 r;
  r.x = ev.x * f + a * xv.x;
  r.y = ev.y * f + a * xv.y;
  r.z = ev.z * f + a * xv.z;
  r.w = ev.w * f + a * xv.w;
  *(f4*)(dst + base) = r;
}

// ---------------------------------------------------------------------------
// Kernel D: stream h-side traces (K=1024). One block (256 thr) per (b,h) row.
// ---------------------------------------------------------------------------
__global__ __launch_bounds__(256) void trace_h(
    const float* __restrict__ h_last,
    const float* __restrict__ e_ih, const float* __restrict__ e_fh,
    const float* __restrict__ e_ch,
    const float* __restrict__ ws, float* __restrict__ out)
{
  const int bh    = blockIdx.x;
  const int which = blockIdx.y;
  const int b     = bh >> 10;

  const float f = ws[WS_GATES + 1 * NBH + bh];
  const float a = ws[WS_COEF + which * NBH + bh];

  const float* e = e_ih; float* dst = out + OFF_EWIH;
  if (which == 1)      { e = e_fh; dst = out + OFF_EWFH; }
  else if (which == 2) { e = e_ch; dst = out + OFF_EWCH; }

  const int base = bh * HID + (threadIdx.x << 2);
  const f4 ev = *(const f4*)(e + base);
  const f4 hv = *(const f4*)(h_last + b * HID + (threadIdx.x << 2));
  f4 r;
  r.x = ev.x * f + a * hv.x;
  r.y = ev.y * f + a * hv.y;
  r.z = ev.z * f + a * hv.z;
  r.w = ev.w * f + a * hv.w;
  *(f4*)(dst + base) = r;
}

extern "C" void kernel_launch(void* const* d_in, const int* in_sizes, int n_in,
                              void* d_out, int out_size, void* d_ws, size_t ws_size,
                              hipStream_t stream) {
  const float* x      = (const float*)d_in[0];
  const float* w_ix   = (const float*)d_in[1];
  const float* w_ih   = (const float*)d_in[2];
  const float* b_i    = (const float*)d_in[3];
  const float* w_fx   = (const float*)d_in[4];
  const float* w_fh   = (const float*)d_in[5];
  const float* b_f    = (const float*)d_in[6];
  const float* w_ox   = (const float*)d_in[7];
  const float* w_oh   = (const float*)d_in[8];
  const float* b_o    = (const float*)d_in[9];
  const float* w_cx   = (const float*)d_in[10];
  const float* w_ch   = (const float*)d_in[11];
  const float* b_c    = (const float*)d_in[12];
  const float* h_last = (const float*)d_in[13];
  const float* c_last = (const float*)d_in[14];
  const float* e_w_ix = (const float*)d_in[15];
  const float* e_w_ih = (const float*)d_in[16];
  const float* e_b_i  = (const float*)d_in[17];
  const float* e_w_fx = (const float*)d_in[18];
  const float* e_w_fh = (const float*)d_in[19];
  const float* e_b_f  = (const float*)d_in[20];
  const float* e_w_cx = (const float*)d_in[21];
  const float* e_w_ch = (const float*)d_in[22];
  const float* e_b_c  = (const float*)d_in[23];

  float* out = (float*)d_out;
  float* ws  = (float*)d_ws;

  // A: 64 H-tiles x 4 gates, one wave each; fp32 WMMA 16x16x4
  gates_wmma<<<dim3(64, 4), 32, 0, stream>>>(
      x, h_last,
      w_ix, w_ih, b_i, w_fx, w_fh, b_f,
      w_ox, w_oh, b_o, w_cx, w_ch, b_c, ws);

  // B: elementwise combine over B*H
  combine<<<dim3(NBH / 256), 256, 0, stream>>>(c_last, e_b_i, e_b_f, e_b_c, ws, out);

  // C: x-side traces (3 tensors, K=512)
  trace_x<<<dim3(NBH, 3), 128, 0, stream>>>(x, e_w_ix, e_w_fx, e_w_cx, ws, out);

  // D: h-side traces (3 tensors, K=1024)
  trace_h<<<dim3(NBH, 3), 256, 0, stream>>>(h_last, e_w_ih, e_w_fh, e_w_ch, ws, out);
}